// LightGCN_28475633172844
// MI455X (gfx1250) — compile-verified
//
#include <hip/hip_runtime.h>
#include <hip/hip_bf16.h>
#include <math.h>

// ---------------------------------------------------------------------------
// LightGCN-style hypergraph conv + classifier + log_softmax for MI455X
//   X0 = concat(node_emb, hyper_emb)            (200000 x 64, fp32)
//   3x: X <- A*X (COO spmm, 3.2M nnz), acc += X
//   Z  = acc @ (W/4) + b                        (WMMA f32 16x16x4, LDS-staged W)
//   out = log_softmax(Z, axis=1)                (fused, in-register shuffles)
// ---------------------------------------------------------------------------

constexpr int   kNodes   = 150000;
constexpr int   kHyper   = 50000;
constexpr int   kTotal   = kNodes + kHyper;      // 200000
constexpr int   kDim     = 64;
constexpr int   kCls     = 50;
constexpr int   kNnz     = 3200000;
constexpr int   kLayers  = 3;
constexpr long long kElems = (long long)kTotal * kDim;   // 12.8M floats

typedef __attribute__((ext_vector_type(2)))  float v2f;
typedef __attribute__((ext_vector_type(8)))  float v8f;

// ---------------------------------------------------------------------------
// init: cur = acc = concat(node_emb, hyper_emb)
// ---------------------------------------------------------------------------
__global__ void lg_init_kernel(const float* __restrict__ ne,
                               const float* __restrict__ he,
                               float* __restrict__ cur,
                               float* __restrict__ acc) {
  long long tid = (long long)blockIdx.x * blockDim.x + threadIdx.x;
  if (tid >= kElems) return;
  const long long nNodeElems = (long long)kNodes * kDim;
  float v = (tid < nNodeElems) ? ne[tid] : he[tid - nNodeElems];
  cur[tid] = v;
  acc[tid] = v;
}

__global__ void lg_zero_kernel(float* __restrict__ p) {
  long long tid = (long long)blockIdx.x * blockDim.x + threadIdx.x;
  if (tid < kElems) p[tid] = 0.0f;
}

__global__ void lg_addacc_kernel(float* __restrict__ acc,
                                 const float* __restrict__ x) {
  long long tid = (long long)blockIdx.x * blockDim.x + threadIdx.x;
  if (tid < kElems) acc[tid] += x[tid];
}

// ---------------------------------------------------------------------------
// COO SpMM: y[rows[e], :] += vals[e] * x[cols[e], :]
// 16 threads per nnz, 4 contiguous dims each (float4 gather, 4x f32 atomics).
// The 51.2MB embedding matrix fits in the 192MB L2, so random gathers and
// non-returning GLOBAL_ATOMIC_ADD_F32 scatters both resolve in L2.
// ---------------------------------------------------------------------------
__global__ void lg_spmm_kernel(const float* __restrict__ vals,
                               const int*   __restrict__ rows,
                               const int*   __restrict__ cols,
                               const float* __restrict__ x,
                               float* __restrict__ y) {
  long long tid = (long long)blockIdx.x * blockDim.x + threadIdx.x;
  const long long total = (long long)kNnz * 16;
  if (tid >= total) return;
  const int e = (int)(tid >> 4);
  const int d = (int)(tid & 15) * 4;
  const float v = vals[e];
  const int   c = cols[e];
  const int   r = rows[e];
  const float4 xv = *(const float4*)(x + (long long)c * kDim + d);
  float* yp = y + (long long)r * kDim + d;
  atomicAdd(yp + 0, v * xv.x);
  atomicAdd(yp + 1, v * xv.y);
  atomicAdd(yp + 2, v * xv.z);
  atomicAdd(yp + 3, v * xv.w);
}

// ---------------------------------------------------------------------------
// Fused classifier: Z = acc @ (W/4) + b, then row-wise log_softmax.
// One wave32 per 16-row stripe; N=50 padded to 4 tiles of 16.
//
// W is staged once per block into LDS, transposed + zero-padded + pre-scaled
// by 0.25: shW[col][kk], row stride 68 floats.  B fragment (kb, kb+1) is then
// one contiguous ds_load_b64; bank = (4*col + kb) mod 64, so the 32 lanes x
// 2 words of each request cover all 64 banks exactly once (conflict-free),
// and the col<50 predication vanishes from the inner loop.
//
// C/D layout (16x16 f32): VGPR r -> rows m0+r (lanes 0-15) / m0+r+8
// (lanes 16-31), lane%16 = column; log_softmax reduces each row with 4
// __shfl_xor steps inside the 16-lane half-groups.
// ---------------------------------------------------------------------------
constexpr int kWStride = 68;   // LDS row stride (floats) for bank-conflict-free B

__global__ void lg_gemm_logsoftmax_kernel(const float* __restrict__ acc,
                                          const float* __restrict__ W,
                                          const float* __restrict__ bias,
                                          float* __restrict__ out) {
  __shared__ float shW[64 * kWStride];  // shW[col*68 + kk] = 0.25*W[kk][col] (padded)
  __shared__ float shB[64];

  for (int idx = threadIdx.x; idx < 64 * 64; idx += blockDim.x) {
    const int col = idx & 63;
    const int kk  = idx >> 6;
    shW[col * kWStride + kk] = (col < kCls) ? 0.25f * W[kk * kCls + col] : 0.0f;
  }
  if (threadIdx.x < 64)
    shB[threadIdx.x] = (threadIdx.x < kCls) ? bias[threadIdx.x] : 0.0f;
  __syncthreads();

  const int wave = blockIdx.x * (blockDim.x >> 5) + (threadIdx.x >> 5);
  const int lane = threadIdx.x & 31;
  const int m0   = wave * 16;
  if (m0 >= kNodes) return;           // uniform per wave: EXEC all-1s below

  const int half = lane >> 4;         // 0: lanes 0-15, 1: lanes 16-31
  const int l15  = lane & 15;

  v8f c[4] = {v8f{}, v8f{}, v8f{}, v8f{}};

  // A layout (32-bit A 16x4): lanes 0-15 hold K=kb,kb+1 with kb=4k;
  // lanes 16-31 hold K=4k+2,4k+3 (per M row).
  const float* arow = acc + (long long)(m0 + l15) * kDim;
  for (int k = 0; k < 16; ++k) {
    const int kb = 4 * k + 2 * half;
    const v2f a = *(const v2f*)(arow + kb);
#pragma unroll
    for (int t = 0; t < 4; ++t) {
      const v2f b = *(const v2f*)(shW + (t * 16 + l15) * kWStride + kb);
      c[t] = __builtin_amdgcn_wmma_f32_16x16x4_f32(
          false, a, false, b, (short)0, c[t], false, false);
    }
  }

  float bv[4];
#pragma unroll
  for (int t = 0; t < 4; ++t) bv[t] = shB[t * 16 + l15];

#pragma unroll
  for (int r = 0; r < 8; ++r) {
    const int row = m0 + r + 8 * half;
    float z[4];
    float m = -1e30f;
#pragma unroll
    for (int t = 0; t < 4; ++t) {
      z[t] = c[t][r] + bv[t];
      if (t * 16 + l15 < kCls) m = fmaxf(m, z[t]);
    }
#pragma unroll
    for (int off = 1; off < 16; off <<= 1)
      m = fmaxf(m, __shfl_xor(m, off, 32));
    float s = 0.0f;
#pragma unroll
    for (int t = 0; t < 4; ++t)
      if (t * 16 + l15 < kCls) s += expf(z[t] - m);
#pragma unroll
    for (int off = 1; off < 16; off <<= 1)
      s += __shfl_xor(s, off, 32);
    const float lse = m + logf(s);
#pragma unroll
    for (int t = 0; t < 4; ++t) {
      const int col = t * 16 + l15;
      if (col < kCls) out[(long long)row * kCls + col] = z[t] - lse;
    }
  }
}

// ---------------------------------------------------------------------------
extern "C" void kernel_launch(void* const* d_in, const int* in_sizes, int n_in,
                              void* d_out, int out_size, void* d_ws, size_t ws_size,
                              hipStream_t stream) {
  const float* node_emb  = (const float*)d_in[0];
  const float* hyper_emb = (const float*)d_in[1];
  const float* W    = (const float*)d_in[2];
  const float* bias = (const float*)d_in[3];
  const float* vals = (const float*)d_in[4];
  const int*   rows = (const int*)d_in[5];
  const int*   cols = (const int*)d_in[6];
  float* out = (float*)d_out;

  float* cur = (float*)d_ws;            // 12.8M floats
  float* nxt = cur + kElems;            // 12.8M floats
  float* acc = nxt + kElems;            // 12.8M floats  (~154 MB total)

  const int elemBlocks = (int)((kElems + 255) / 256);
  lg_init_kernel<<<elemBlocks, 256, 0, stream>>>(node_emb, hyper_emb, cur, acc);

  const long long spmmThreads = (long long)kNnz * 16;
  const int spmmBlocks = (int)((spmmThreads + 255) / 256);
  for (int l = 0; l < kLayers; ++l) {
    lg_zero_kernel<<<elemBlocks, 256, 0, stream>>>(nxt);
    lg_spmm_kernel<<<spmmBlocks, 256, 0, stream>>>(vals, rows, cols, cur, nxt);
    lg_addacc_kernel<<<elemBlocks, 256, 0, stream>>>(acc, nxt);
    float* tmp = cur; cur = nxt; nxt = tmp;   // deterministic host-side swap
  }

  // 9375 row-stripes of 16, 8 waves per 256-thread block
  const int nWaves  = (kNodes + 15) / 16;
  const int nBlocks = (nWaves + 7) / 8;
  lg_gemm_logsoftmax_kernel<<<nBlocks, 256, 0, stream>>>(acc, W, bias, out);
}